// DreamGaussianModel_61263413510371
// MI455X (gfx1250) — compile-verified
//
#include <hip/hip_runtime.h>
#include <hip/hip_bf16.h>

typedef __attribute__((ext_vector_type(2))) float v2f;
typedef __attribute__((ext_vector_type(8))) float v8f;
typedef __attribute__((ext_vector_type(4))) int v4i;
typedef unsigned long long u64;

#define IMG_W 512
#define IMG_H 512
#define PADV 2
#define FXc 409.6f
#define FYc 409.6f
#define CXc 256.0f
#define CYc 256.0f

#if defined(__has_builtin)
#if __has_builtin(__builtin_amdgcn_global_load_async_to_lds_b128)
#define HAVE_ASYNC_LDS 1
#endif
#endif
#ifndef HAVE_ASYNC_LDS
#define HAVE_ASYNC_LDS 0
#endif

#if defined(__has_builtin) && __has_builtin(__builtin_amdgcn_s_wait_asynccnt)
#define WAIT_ASYNC(n) __builtin_amdgcn_s_wait_asynccnt(n)
#else
#define WAIT_ASYNC(n) asm volatile("s_wait_asynccnt %0" ::"i"(n))
#endif

#if HAVE_ASYNC_LDS
// b128 async copy global->LDS; builtin takes AS(1)/AS(3) v4i pointers (per clang diagnostic).
__device__ __forceinline__ void async_copy16(const void* g, void* l) {
    __builtin_amdgcn_global_load_async_to_lds_b128(
        (__attribute__((address_space(1))) v4i*)(void*)g,
        (__attribute__((address_space(3))) v4i*)(void*)l, 0, 0);
}
#endif

// exp(-(dx^2+dy^2)/2) for dx,dy in [-2,2]
__device__ __constant__ float cW5[25] = {
    0.0183156393f, 0.0820849985f, 0.1353352832f, 0.0820849985f, 0.0183156393f,
    0.0820849985f, 0.3678794412f, 0.6065306597f, 0.3678794412f, 0.0820849985f,
    0.1353352832f, 0.6065306597f, 1.0000000000f, 0.6065306597f, 0.1353352832f,
    0.0820849985f, 0.3678794412f, 0.6065306597f, 0.3678794412f, 0.0820849985f,
    0.0183156393f, 0.0820849985f, 0.1353352832f, 0.0820849985f, 0.0183156393f};

// ---------------- 4x4 inverse (adjugate), 1 thread ----------------
__global__ void pose_inverse_kernel(const float* __restrict__ P, float* __restrict__ Pinv) {
    if (threadIdx.x != 0 || blockIdx.x != 0) return;
    float m[16];
#pragma unroll
    for (int i = 0; i < 16; ++i) m[i] = P[i];
    float inv[16];
    inv[0]  =  m[5]*m[10]*m[15] - m[5]*m[11]*m[14] - m[9]*m[6]*m[15] + m[9]*m[7]*m[14] + m[13]*m[6]*m[11] - m[13]*m[7]*m[10];
    inv[4]  = -m[4]*m[10]*m[15] + m[4]*m[11]*m[14] + m[8]*m[6]*m[15] - m[8]*m[7]*m[14] - m[12]*m[6]*m[11] + m[12]*m[7]*m[10];
    inv[8]  =  m[4]*m[9]*m[15]  - m[4]*m[11]*m[13] - m[8]*m[5]*m[15] + m[8]*m[7]*m[13] + m[12]*m[5]*m[11] - m[12]*m[7]*m[9];
    inv[12] = -m[4]*m[9]*m[14]  + m[4]*m[10]*m[13] + m[8]*m[5]*m[14] - m[8]*m[6]*m[13] - m[12]*m[5]*m[10] + m[12]*m[6]*m[9];
    inv[1]  = -m[1]*m[10]*m[15] + m[1]*m[11]*m[14] + m[9]*m[2]*m[15] - m[9]*m[3]*m[14] - m[13]*m[2]*m[11] + m[13]*m[3]*m[10];
    inv[5]  =  m[0]*m[10]*m[15] - m[0]*m[11]*m[14] - m[8]*m[2]*m[15] + m[8]*m[3]*m[14] + m[12]*m[2]*m[11] - m[12]*m[3]*m[10];
    inv[9]  = -m[0]*m[9]*m[15]  + m[0]*m[11]*m[13] + m[8]*m[1]*m[15] - m[8]*m[3]*m[13] - m[12]*m[1]*m[11] + m[12]*m[3]*m[9];
    inv[13] =  m[0]*m[9]*m[14]  - m[0]*m[10]*m[13] - m[8]*m[1]*m[14] + m[8]*m[2]*m[13] + m[12]*m[1]*m[10] - m[12]*m[2]*m[9];
    inv[2]  =  m[1]*m[6]*m[15]  - m[1]*m[7]*m[14]  - m[5]*m[2]*m[15] + m[5]*m[3]*m[14] + m[13]*m[2]*m[7]  - m[13]*m[3]*m[6];
    inv[6]  = -m[0]*m[6]*m[15]  + m[0]*m[7]*m[14]  + m[4]*m[2]*m[15] - m[4]*m[3]*m[14] - m[12]*m[2]*m[7]  + m[12]*m[3]*m[6];
    inv[10] =  m[0]*m[5]*m[15]  - m[0]*m[7]*m[13]  - m[4]*m[1]*m[15] + m[4]*m[3]*m[13] + m[12]*m[1]*m[7]  - m[12]*m[3]*m[5];
    inv[14] = -m[0]*m[5]*m[14]  + m[0]*m[6]*m[13]  + m[4]*m[1]*m[14] - m[4]*m[2]*m[13] - m[12]*m[1]*m[6]  + m[12]*m[2]*m[5];
    inv[3]  = -m[1]*m[6]*m[11]  + m[1]*m[7]*m[10]  + m[5]*m[2]*m[11] - m[5]*m[3]*m[10] - m[9]*m[2]*m[7]   + m[9]*m[3]*m[6];
    inv[7]  =  m[0]*m[6]*m[11]  - m[0]*m[7]*m[10]  - m[4]*m[2]*m[11] + m[4]*m[3]*m[10] + m[8]*m[2]*m[7]   - m[8]*m[3]*m[6];
    inv[11] = -m[0]*m[5]*m[11]  + m[0]*m[7]*m[9]   + m[4]*m[1]*m[11] - m[4]*m[3]*m[9]  - m[8]*m[1]*m[7]   + m[8]*m[3]*m[5];
    inv[15] =  m[0]*m[5]*m[10]  - m[0]*m[6]*m[9]   - m[4]*m[1]*m[10] + m[4]*m[2]*m[9]  + m[8]*m[1]*m[6]   - m[8]*m[2]*m[5];
    float det = m[0]*inv[0] + m[1]*inv[4] + m[2]*inv[8] + m[3]*inv[12];
    float rdet = 1.0f / det;
#pragma unroll
    for (int i = 0; i < 16; ++i) Pinv[i] = inv[i] * rdet;
}

__device__ __forceinline__ float sigmoidf(float x) { return 1.0f / (1.0f + expf(-x)); }

// ------------- WMMA projection: 16 gaussians per wave32 -------------
// D = A(16x4 = padded pose_inv) x B(4x16 = homo^T) ; D[m][n] = cam comp m of gaussian n
__global__ __launch_bounds__(256) void project_kernel(
    const float* __restrict__ Pinv, const float* __restrict__ means,
    const float* __restrict__ sh, const float* __restrict__ opac,
    const unsigned char* __restrict__ mask,
    int* __restrict__ xiA, int* __restrict__ yiA,
    float* __restrict__ aA, float* __restrict__ rA, float* __restrict__ gA, float* __restrict__ bA,
    u64* __restrict__ keys, int N) {
    int wid  = (blockIdx.x * blockDim.x + threadIdx.x) >> 5;
    int lane = threadIdx.x & 31;
    int col  = lane & 15;
    int half = lane >> 4;
    int g    = wid * 16 + col;
    int gc   = g < N ? g : N - 1;

    // Unconditional loads (clamped) + selects: keep EXEC uniform into the WMMA.
    // A: 16x4 f32 A-matrix layout. VGPR0: K=0 (lanes 0-15) / K=2 (lanes 16-31); VGPR1: K=1 / K=3.
    float p0 = Pinv[(col & 3) * 4 + (half ? 2 : 0)];
    float p1 = Pinv[(col & 3) * 4 + (half ? 3 : 1)];
    v2f A;
    A.x = (col < 4) ? p0 : 0.0f;
    A.y = (col < 4) ? p1 : 0.0f;
    // B: 4x16 f32 B-matrix layout. VGPR0: row K=0 / K=2; VGPR1: row K=1 / K=3. B[k][n] = homo[n][k].
    const float* mg = means + (size_t)gc * 3;
    float m0 = mg[0], m1 = mg[1], m2 = mg[2];
    v2f B;
    B.x = half ? m2 : m0;
    B.y = half ? 1.0f : m1;

    v8f C = {};
    v8f D = __builtin_amdgcn_wmma_f32_16x16x4_f32(false, A, false, B, (short)0, C,
                                                  false, false);
    if (half == 0 && g < N) {
        float cx = D[0], cy = D[1], cz = D[2];
        float rz = 1.0f / cz;  // one IEEE division, reused for both axes
        float x2d = fmaf(cx * FXc, rz, CXc);
        float y2d = fmaf(cy * FYc, rz, CYc);
        float tx = truncf(x2d), ty = truncf(y2d);
        bool valid = (mask[g] != 0) && (tx >= 0.0f) && (tx < (float)IMG_W) &&
                     (ty >= 0.0f) && (ty < (float)IMG_H);
        float xc = fminf(fmaxf(tx + (float)PADV, (float)PADV), (float)(IMG_W + PADV - 1));
        float yc = fminf(fmaxf(ty + (float)PADV, (float)PADV), (float)(IMG_H + PADV - 1));
        xiA[g] = (int)xc;
        yiA[g] = (int)yc;
        aA[g] = opac[g] * (valid ? 1.0f : 0.0f);
        const float* shg = sh + (size_t)g * 48;
        rA[g] = sigmoidf(shg[0]);
        gA[g] = sigmoidf(shg[16]);
        bA[g] = sigmoidf(shg[32]);
        // stable descending-z key: sort ascending on (~orderable(z) , index)
        unsigned bits = __float_as_uint(cz);
        unsigned u = (bits & 0x80000000u) ? ~bits : (bits | 0x80000000u);
        keys[g] = ((u64)(~u) << 32) | (unsigned)g;
    }
}

// ---------------- bitonic sort (ascending) on u64 keys ----------------
__global__ __launch_bounds__(256) void bitonic_local_sort(u64* __restrict__ keys) {
    __shared__ u64 sk[256];
    int t = threadIdx.x;
    int gbase = blockIdx.x * 256;
    sk[t] = keys[gbase + t];
    __syncthreads();
    for (int k = 2; k <= 256; k <<= 1) {
        for (int j = k >> 1; j > 0; j >>= 1) {
            int l = t ^ j;
            if (l > t) {
                bool asc = (((gbase + t) & k) == 0);
                u64 a = sk[t], b = sk[l];
                if ((a > b) == asc) { sk[t] = b; sk[l] = a; }
            }
            __syncthreads();
        }
    }
    keys[gbase + t] = sk[t];
}

__global__ __launch_bounds__(256) void bitonic_global_pass(u64* __restrict__ keys, int k, int j) {
    int i = blockIdx.x * blockDim.x + threadIdx.x;
    int l = i ^ j;
    if (l > i) {
        u64 a = keys[i], b = keys[l];
        bool asc = ((i & k) == 0);
        if ((a > b) == asc) { keys[i] = b; keys[l] = a; }
    }
}

__global__ __launch_bounds__(256) void bitonic_local_merge(u64* __restrict__ keys, int k) {
    __shared__ u64 sk[256];
    int t = threadIdx.x;
    int gbase = blockIdx.x * 256;
    sk[t] = keys[gbase + t];
    bool asc = (((gbase + t) & k) == 0); // k >= 512: uniform per block
    __syncthreads();
    for (int j = 128; j > 0; j >>= 1) {
        int l = t ^ j;
        if (l > t) {
            u64 a = sk[t], b = sk[l];
            if ((a > b) == asc) { sk[t] = b; sk[l] = a; }
        }
        __syncthreads();
    }
    keys[gbase + t] = sk[t];
}

// ------------- gather attributes into depth order (coalesced AoS) -------------
__global__ __launch_bounds__(256) void reorder_kernel(
    const u64* __restrict__ keys, const int* __restrict__ xiA, const int* __restrict__ yiA,
    const float* __restrict__ aA, const float* __restrict__ rA, const float* __restrict__ gA,
    const float* __restrict__ bA, float4* __restrict__ sorted, int N) {
    int s = blockIdx.x * blockDim.x + threadIdx.x;
    if (s >= N) return;
    unsigned g = (unsigned)keys[s];
    float4 e0 = make_float4(__int_as_float(xiA[g]), __int_as_float(yiA[g]), aA[g], rA[g]);
    float4 e1 = make_float4(gA[g], bA[g], 0.0f, 0.0f);
    sorted[2 * s + 0] = e0;
    sorted[2 * s + 1] = e1;
}

// ------------- per-tile ordered compositing (16x16 pixels per block) -------------
// Double-buffered async-to-LDS streaming of the depth-sorted gaussian list.
__global__ __launch_bounds__(256) void blend_kernel(const float4* __restrict__ sorted,
                                                    float* __restrict__ out, int N) {
    __shared__ int   s_xy[256];
    __shared__ float s_a[256], s_r[256], s_g[256], s_b[256];
    __shared__ int   s_wtot[8];
#if HAVE_ASYNC_LDS
    __shared__ alignas(16) char stage[2][8192];  // 256 entries x 32 B, double buffered
#endif

    int tile = blockIdx.x;
    int tx0 = (tile & 31) * 16;
    int ty0 = (tile >> 5) * 16;
    int lx = threadIdx.x & 15, ly = threadIdx.x >> 4;
    int px = tx0 + lx + PADV;  // padded coords
    int py = ty0 + ly + PADV;
    int lane = threadIdx.x & 31, wave = threadIdx.x >> 5;

    float cR = 0.0f, cG = 0.0f, cB = 0.0f;
    int nbatch = N / 256;

#if HAVE_ASYNC_LDS
    const char* gbase = (const char*)sorted;
    // prologue: async-load batch 0 into buffer 0 (each thread: 2 x B128 lanes)
    {
        const char* src = gbase + (size_t)threadIdx.x * 16;
        async_copy16(src, &stage[0][threadIdx.x * 16]);
        async_copy16(src + 4096, &stage[0][4096 + threadIdx.x * 16]);
    }
#endif

    for (int b = 0; b < nbatch; ++b) {
        float4 e0, e1;
#if HAVE_ASYNC_LDS
        int buf = b & 1;
        if (b + 1 < nbatch) {
            const char* src = gbase + (size_t)(b + 1) * 8192 + (size_t)threadIdx.x * 16;
            async_copy16(src, &stage[buf ^ 1][threadIdx.x * 16]);
            async_copy16(src + 4096, &stage[buf ^ 1][4096 + threadIdx.x * 16]);
            WAIT_ASYNC(2);  // async loads complete in order: batch b is resident
        } else {
            WAIT_ASYNC(0);
        }
        __syncthreads();
        const float4* ent = (const float4*)&stage[buf][threadIdx.x * 32];
        e0 = ent[0];
        e1 = ent[1];
#else
        int i = b * 256 + threadIdx.x;
        e0 = sorted[2 * i + 0];
        e1 = sorted[2 * i + 1];
        if (b + 1 < nbatch) __builtin_prefetch(&sorted[2 * (i + 256)], 0, 0);
#endif
        int xi = __float_as_int(e0.x);
        int yi = __float_as_int(e0.y);
        float a = e0.z;
        // gaussian 5x5 footprint overlaps this tile's padded window?
        bool pred = (a > 0.0f) && (xi >= tx0) && (xi <= tx0 + 19) && (yi >= ty0) && (yi <= ty0 + 19);

        // order-preserving block compaction (wave32 ballot + cross-wave LDS prefix)
        unsigned long long bal = __ballot((int)pred);
        int prefix = __popcll(bal & ((1ull << lane) - 1ull));
        if (lane == 0) s_wtot[wave] = (int)__popcll(bal);
        __syncthreads();
        int woff = 0, total = 0;
#pragma unroll
        for (int wv = 0; wv < 8; ++wv) {
            int c = s_wtot[wv];
            if (wv < wave) woff += c;
            total += c;
        }
        if (pred) {
            int idx = woff + prefix;
            s_xy[idx] = (yi << 16) | xi;
            s_a[idx] = a;
            s_r[idx] = e0.w;
            s_g[idx] = e1.x;
            s_b[idx] = e1.y;
        }
        __syncthreads();

        for (int c = 0; c < total; ++c) {
            int xy = s_xy[c];
            int dx = px - (xy & 0xFFFF) + 2;
            int dy = py - (xy >> 16) + 2;
            if ((unsigned)dx <= 4u && (unsigned)dy <= 4u) {
                float aw = s_a[c] * cW5[dy * 5 + dx];
                cR = fmaf(aw, s_r[c] - cR, cR);
                cG = fmaf(aw, s_g[c] - cG, cG);
                cB = fmaf(aw, s_b[c] - cB, cB);
            }
        }
        __syncthreads();
    }

    int ox = tx0 + lx, oy = ty0 + ly;
    int p = oy * IMG_W + ox;
    out[0 * IMG_W * IMG_H + p] = cR;
    out[1 * IMG_W * IMG_H + p] = cG;
    out[2 * IMG_W * IMG_H + p] = cB;
}

extern "C" void kernel_launch(void* const* d_in, const int* in_sizes, int n_in,
                              void* d_out, int out_size, void* d_ws, size_t ws_size,
                              hipStream_t stream) {
    const float* pose  = (const float*)d_in[0];
    const float* means = (const float*)d_in[1];
    const float* sh    = (const float*)d_in[2];
    const float* opac  = (const float*)d_in[3];
    const unsigned char* mask = (const unsigned char*)d_in[4];
    float* out = (float*)d_out;
    const int N = in_sizes[1] / 3;  // 262144 (power of two)

    char* ws = (char*)d_ws;
    float* Pinv   = (float*)(ws);                       // 64 B
    u64*   keys   = (u64*)(ws + 64);                    // 8N
    int*   xiA    = (int*)(ws + 64 + 8ull * N);         // 4N
    int*   yiA    = (int*)(ws + 64 + 12ull * N);
    float* aA     = (float*)(ws + 64 + 16ull * N);
    float* rA     = (float*)(ws + 64 + 20ull * N);
    float* gA     = (float*)(ws + 64 + 24ull * N);
    float* bA     = (float*)(ws + 64 + 28ull * N);
    float4* sortedA = (float4*)(ws + 64 + 32ull * N);   // 32N

    pose_inverse_kernel<<<1, 64, 0, stream>>>(pose, Pinv);

    int waves = N / 16;
    int pblocks = (waves * 32 + 255) / 256;
    project_kernel<<<pblocks, 256, 0, stream>>>(Pinv, means, sh, opac, mask,
                                                xiA, yiA, aA, rA, gA, bA, keys, N);

    int sblocks = N / 256;
    bitonic_local_sort<<<sblocks, 256, 0, stream>>>(keys);
    for (int k = 512; k <= N; k <<= 1) {
        for (int j = k >> 1; j >= 256; j >>= 1)
            bitonic_global_pass<<<sblocks, 256, 0, stream>>>(keys, k, j);
        bitonic_local_merge<<<sblocks, 256, 0, stream>>>(keys, k);
    }

    reorder_kernel<<<sblocks, 256, 0, stream>>>(keys, xiA, yiA, aA, rA, gA, bA, sortedA, N);

    blend_kernel<<<(IMG_W / 16) * (IMG_H / 16), 256, 0, stream>>>(sortedA, out, N);
}